// DQNAgent_53154515255692
// MI455X (gfx1250) — compile-verified
//
#include <hip/hip_runtime.h>
#include <hip/hip_bf16.h>
#include <math.h>

typedef __attribute__((ext_vector_type(2))) float v2f;
typedef __attribute__((ext_vector_type(8))) float v8f;

#define BB 16
#define NN 256
#define DD 64
#define HH 64
#define NPAIR 32640      // 256*255/2
#define NTILES 2040      // NPAIR / 16
#define ROWSTRIDE 68     // padded LDS row stride (68 mod 64 = 4 -> conflict-free)

__device__ __forceinline__ int pair_offset(int i) {
  // number of pairs before first-index i (triu, k=1): i*255 - i*(i-1)/2
  return i * 255 - (i * (i - 1)) / 2;
}

// ---------------------------------------------------------------------------
// Pa = O @ W1[0:64,:]  + b1   (Oi segment, bias folded in)
// Pb = O @ W1[64:128,:]       (Oj segment)
// Shapes: O (B*N, 64), W1 (256, 64), Pa/Pb (B*N, 64). 67 MFLOP total.
// ---------------------------------------------------------------------------
__global__ __launch_bounds__(256)
void precompute_pab_kernel(const float* __restrict__ O,
                           const float* __restrict__ W1,
                           const float* __restrict__ b1,
                           float* __restrict__ Pa,
                           float* __restrict__ Pb) {
  const int g = blockIdx.x * blockDim.x + threadIdx.x;   // (bn, h) flat
  if (g >= BB * NN * HH) return;
  const int h  = g & (HH - 1);
  const int bn = g >> 6;
  const float* orow = O + (size_t)bn * DD;
  float sa = b1[h];
  float sb = 0.0f;
  #pragma unroll 8
  for (int k = 0; k < DD; ++k) {
    const float ov = orow[k];
    sa = fmaf(ov, W1[k * HH + h], sa);
    sb = fmaf(ov, W1[(DD + k) * HH + h], sb);
  }
  Pa[g] = sa;
  Pb[g] = sb;
}

// ---------------------------------------------------------------------------
// Shared WMMA step macro: A frag (16x4 f32; lanes 0-15 K=k0,k0+1, lanes 16-31
// K=k0+2,k0+3), B frags (4x16 f32; lanes 0-15 rows k0,k0+1, lanes 16-31 rows
// k0+2,k0+3) read from cache-resident W1.
// ---------------------------------------------------------------------------
#define DO_STEP(SEG, A0, A1)                                                    \
  {                                                                             \
    v2f Af; Af.x = (A0); Af.y = (A1);                                           \
    const int kr0 = (SEG) * 64 + kk * 4 + ksel;                                 \
    const float* wp0 = W1 + kr0 * HH + ncol;                                    \
    const float* wp1 = wp0 + HH;                                                \
    v2f B0; B0.x = wp0[0];  B0.y = wp1[0];                                      \
    v2f B1; B1.x = wp0[16]; B1.y = wp1[16];                                     \
    v2f B2; B2.x = wp0[32]; B2.y = wp1[32];                                     \
    v2f B3; B3.x = wp0[48]; B3.y = wp1[48];                                     \
    acc0 = __builtin_amdgcn_wmma_f32_16x16x4_f32(false, Af, false, B0,          \
                                                 (short)0, acc0, false, false); \
    acc1 = __builtin_amdgcn_wmma_f32_16x16x4_f32(false, Af, false, B1,          \
                                                 (short)0, acc1, false, false); \
    acc2 = __builtin_amdgcn_wmma_f32_16x16x4_f32(false, Af, false, B2,          \
                                                 (short)0, acc2, false, false); \
    acc3 = __builtin_amdgcn_wmma_f32_16x16x4_f32(false, Af, false, B3,          \
                                                 (short)0, acc3, false, false); \
  }

// ---------------------------------------------------------------------------
// Common per-tile prologue: pair-index solve + LDS staging of Oi/Oj rows.
// ---------------------------------------------------------------------------
#define TILE_PROLOGUE()                                                         \
  const int tile = blockIdx.x;                                                  \
  const int b    = blockIdx.y;                                                  \
  const int lane = threadIdx.x;                                                 \
  if (lane < 16) {                                                              \
    const int m = tile * 16 + lane;                                             \
    float fi = 255.5f - sqrtf(255.5f * 255.5f - 2.0f * (float)m);               \
    int i = (int)fi;                                                            \
    if (i < 0) i = 0;                                                           \
    if (i > 254) i = 254;                                                       \
    while (i < 254 && pair_offset(i + 1) <= m) ++i;                             \
    while (i > 0 && pair_offset(i) > m) --i;                                    \
    const int j = i + 1 + (m - pair_offset(i));                                 \
    sIdxI[lane] = i;                                                            \
    sIdxJ[lane] = j;                                                            \
  }                                                                             \
  __syncthreads();                                                              \
  const float* Ob = O + (size_t)b * NN * DD;                                    \
  _Pragma("unroll")                                                             \
  for (int it = 0; it < 8; ++it) {                                              \
    const int flat = it * 32 + lane;                                            \
    const int r    = flat >> 4;                                                 \
    const int c4   = (flat & 15) << 2;                                          \
    const int ri = sIdxI[r];                                                    \
    const int rj = sIdxJ[r];                                                    \
    const float4 vi = *(const float4*)(Ob + ri * DD + c4);                      \
    const float4 vj = *(const float4*)(Ob + rj * DD + c4);                      \
    *(float4*)(&ldsI[r * ROWSTRIDE + c4]) = vi;                                 \
    *(float4*)(&ldsJ[r * ROWSTRIDE + c4]) = vj;                                 \
  }                                                                             \
  __syncthreads();                                                              \
  const int row  = lane & 15;                                                   \
  const int ncol = lane & 15;                                                   \
  const int ksel = (lane >> 4) * 2;                                             \
  const float* rowI = &ldsI[row * ROWSTRIDE];                                   \
  const float* rowJ = &ldsJ[row * ROWSTRIDE];

// ---------------------------------------------------------------------------
// Common epilogue: Q = relu(h) @ W2 + b2, lane-column reduction, store.
// ---------------------------------------------------------------------------
#define TILE_EPILOGUE()                                                         \
  const float w20 = W2[ncol];                                                   \
  const float w21 = W2[16 + ncol];                                              \
  const float w22 = W2[32 + ncol];                                              \
  const float w23 = W2[48 + ncol];                                              \
  const float bq  = b2[0];                                                      \
  float qp[8];                                                                  \
  _Pragma("unroll")                                                             \
  for (int g = 0; g < 8; ++g) {                                                 \
    qp[g] = fmaxf(acc0[g], 0.0f) * w20 + fmaxf(acc1[g], 0.0f) * w21 +           \
            fmaxf(acc2[g], 0.0f) * w22 + fmaxf(acc3[g], 0.0f) * w23;            \
  }                                                                             \
  _Pragma("unroll")                                                             \
  for (int mk = 1; mk <= 8; mk <<= 1) {                                         \
    _Pragma("unroll")                                                           \
    for (int g = 0; g < 8; ++g) qp[g] += __shfl_xor(qp[g], mk, 16);             \
  }                                                                             \
  if ((lane & 15) == 0) {                                                       \
    const int rbase2 = (lane >> 4) * 8;                                         \
    float* qdst = Qout + (size_t)b * NPAIR + tile * 16 + rbase2;                \
    _Pragma("unroll")                                                           \
    for (int g = 0; g < 8; ++g) qdst[g] = qp[g] + bq;                           \
  }

// ---------------------------------------------------------------------------
// Optimized kernel: accumulators seeded with Pa[i] + Pb[j]; K-loop only covers
// the nonlinear segments |Oi-Oj| (rows 128..191) and Oi*Oj (rows 192..255).
// ---------------------------------------------------------------------------
__global__ __launch_bounds__(32)
void dqn_q_split_kernel(const float* __restrict__ O,
                        const float* __restrict__ W1,
                        const float* __restrict__ W2,
                        const float* __restrict__ b2,
                        const float* __restrict__ Pa,
                        const float* __restrict__ Pb,
                        float* __restrict__ Qout) {
  __shared__ float ldsI[16 * ROWSTRIDE];
  __shared__ float ldsJ[16 * ROWSTRIDE];
  __shared__ int sIdxI[16];
  __shared__ int sIdxJ[16];

  TILE_PROLOGUE();

  // Seed C fragments with Pa[i_row] + Pb[j_row] (b1 already folded into Pa).
  const float* PaB = Pa + (size_t)b * NN * HH;
  const float* PbB = Pb + (size_t)b * NN * HH;
  const int rbase = (lane >> 4) * 8;   // lanes 0-15: rows 0-7, lanes 16-31: 8-15
  v8f acc0, acc1, acc2, acc3;
  #pragma unroll
  for (int g = 0; g < 8; ++g) {
    const int ri = sIdxI[rbase + g];
    const int rj = sIdxJ[rbase + g];
    const float* pa = PaB + ri * HH + ncol;
    const float* pb = PbB + rj * HH + ncol;
    acc0[g] = pa[0]  + pb[0];
    acc1[g] = pa[16] + pb[16];
    acc2[g] = pa[32] + pb[32];
    acc3[g] = pa[48] + pb[48];
  }

  #pragma unroll 4
  for (int kk = 0; kk < 16; ++kk) {           // segment 2: |Oi - Oj|
    const int c = kk * 4 + ksel;
    DO_STEP(2, fabsf(rowI[c] - rowJ[c]), fabsf(rowI[c + 1] - rowJ[c + 1]));
  }
  #pragma unroll 4
  for (int kk = 0; kk < 16; ++kk) {           // segment 3: Oi * Oj
    const int c = kk * 4 + ksel;
    DO_STEP(3, rowI[c] * rowJ[c], rowI[c + 1] * rowJ[c + 1]);
  }

  TILE_EPILOGUE();
}

// ---------------------------------------------------------------------------
// Fallback kernel (used only if d_ws is too small for Pa/Pb): full K = 256.
// ---------------------------------------------------------------------------
__global__ __launch_bounds__(32)
void dqn_q_full_kernel(const float* __restrict__ O,
                       const float* __restrict__ W1,
                       const float* __restrict__ b1,
                       const float* __restrict__ W2,
                       const float* __restrict__ b2,
                       float* __restrict__ Qout) {
  __shared__ float ldsI[16 * ROWSTRIDE];
  __shared__ float ldsJ[16 * ROWSTRIDE];
  __shared__ int sIdxI[16];
  __shared__ int sIdxJ[16];

  TILE_PROLOGUE();

  const float bb0 = b1[ncol];
  const float bb1 = b1[16 + ncol];
  const float bb2 = b1[32 + ncol];
  const float bb3 = b1[48 + ncol];
  v8f acc0 = {bb0, bb0, bb0, bb0, bb0, bb0, bb0, bb0};
  v8f acc1 = {bb1, bb1, bb1, bb1, bb1, bb1, bb1, bb1};
  v8f acc2 = {bb2, bb2, bb2, bb2, bb2, bb2, bb2, bb2};
  v8f acc3 = {bb3, bb3, bb3, bb3, bb3, bb3, bb3, bb3};

  #pragma unroll 4
  for (int kk = 0; kk < 16; ++kk) {           // segment 0: Oi
    const int c = kk * 4 + ksel;
    DO_STEP(0, rowI[c], rowI[c + 1]);
  }
  #pragma unroll 4
  for (int kk = 0; kk < 16; ++kk) {           // segment 1: Oj
    const int c = kk * 4 + ksel;
    DO_STEP(1, rowJ[c], rowJ[c + 1]);
  }
  #pragma unroll 4
  for (int kk = 0; kk < 16; ++kk) {           // segment 2: |Oi - Oj|
    const int c = kk * 4 + ksel;
    DO_STEP(2, fabsf(rowI[c] - rowJ[c]), fabsf(rowI[c + 1] - rowJ[c + 1]));
  }
  #pragma unroll 4
  for (int kk = 0; kk < 16; ++kk) {           // segment 3: Oi * Oj
    const int c = kk * 4 + ksel;
    DO_STEP(3, rowI[c] * rowJ[c], rowI[c + 1] * rowJ[c + 1]);
  }

  TILE_EPILOGUE();
}

// ---- Top-2 per batch (first-occurrence tie-break like lax.top_k) ----
__global__ __launch_bounds__(256)
void topk2_kernel(const float* __restrict__ Q, int* __restrict__ idxOut) {
  const int b = blockIdx.x;
  const int tid = threadIdx.x;

  float v0 = -__builtin_inff(), v1 = -__builtin_inff();
  int i0 = 0x7fffffff, i1 = 0x7fffffff;
  for (int m = tid; m < NPAIR; m += 256) {
    const float v = Q[(size_t)b * NPAIR + m];
    if (v > v0 || (v == v0 && m < i0)) { v1 = v0; i1 = i0; v0 = v; i0 = m; }
    else if (v > v1 || (v == v1 && m < i1)) { v1 = v; i1 = m; }
  }

  __shared__ float sv[256][2];
  __shared__ int   si[256][2];
  sv[tid][0] = v0; sv[tid][1] = v1;
  si[tid][0] = i0; si[tid][1] = i1;
  __syncthreads();

  for (int s = 128; s > 0; s >>= 1) {
    if (tid < s) {
      float a0 = sv[tid][0], a1 = sv[tid][1];
      int  ai0 = si[tid][0], ai1 = si[tid][1];
      float c0 = sv[tid + s][0], c1 = sv[tid + s][1];
      int  ci0 = si[tid + s][0], ci1 = si[tid + s][1];
      float r0, r1; int ri0, ri1;
      const bool aFirst = (a0 > c0) || (a0 == c0 && ai0 < ci0);
      if (aFirst) {
        r0 = a0; ri0 = ai0;
        if ((a1 > c0) || (a1 == c0 && ai1 < ci0)) { r1 = a1; ri1 = ai1; }
        else                                      { r1 = c0; ri1 = ci0; }
      } else {
        r0 = c0; ri0 = ci0;
        if ((c1 > a0) || (c1 == a0 && ci1 < ai0)) { r1 = c1; ri1 = ci1; }
        else                                      { r1 = a0; ri1 = ai0; }
      }
      sv[tid][0] = r0; sv[tid][1] = r1;
      si[tid][0] = ri0; si[tid][1] = ri1;
    }
    __syncthreads();
  }
  if (tid == 0) {
    idxOut[b * 2 + 0] = si[0][0];
    idxOut[b * 2 + 1] = si[0][1];
  }
}

// ---- Mask writer ----
__global__ __launch_bounds__(256)
void mask_kernel(const int* __restrict__ idx, float* __restrict__ maskOut) {
  const int g = blockIdx.x * blockDim.x + threadIdx.x;
  if (g >= BB * NPAIR) return;
  const int b = g / NPAIR;
  const int m = g - b * NPAIR;
  maskOut[g] = (m == idx[b * 2] || m == idx[b * 2 + 1]) ? 1.0f : 0.0f;
}

extern "C" void kernel_launch(void* const* d_in, const int* in_sizes, int n_in,
                              void* d_out, int out_size, void* d_ws, size_t ws_size,
                              hipStream_t stream) {
  (void)in_sizes; (void)n_in; (void)out_size;
  const float* O  = (const float*)d_in[0];
  const float* W1 = (const float*)d_in[1];
  const float* b1 = (const float*)d_in[2];
  const float* W2 = (const float*)d_in[3];
  const float* b2 = (const float*)d_in[4];

  float* Qout    = (float*)d_out;                 // B*NPAIR floats
  float* maskOut = Qout + (size_t)BB * NPAIR;     // B*NPAIR floats (0/1)

  const size_t PAB  = (size_t)BB * NN * HH;       // 262144 elements
  const size_t need = 2 * PAB * sizeof(float) + 2 * BB * sizeof(int);

  dim3 grid1(NTILES, BB);
  int* idxWs;

  if (ws_size >= need) {
    // Optimized path: precompute linear-segment partials, halve the K loop.
    float* Pa = (float*)d_ws;
    float* Pb = Pa + PAB;
    idxWs     = (int*)(Pb + PAB);
    const int tot = BB * NN * HH;
    precompute_pab_kernel<<<(tot + 255) / 256, 256, 0, stream>>>(O, W1, b1, Pa, Pb);
    dqn_q_split_kernel<<<grid1, 32, 0, stream>>>(O, W1, W2, b2, Pa, Pb, Qout);
  } else {
    // Fallback: full K=256 WMMA pipeline, only needs 128B of workspace.
    idxWs = (int*)d_ws;
    dqn_q_full_kernel<<<grid1, 32, 0, stream>>>(O, W1, b1, W2, b2, Qout);
  }

  topk2_kernel<<<BB, 256, 0, stream>>>(Qout, idxWs);
  const int total = BB * NPAIR;
  mask_kernel<<<(total + 255) / 256, 256, 0, stream>>>(idxWs, maskOut);
}